// FMFMNeuronBank_42262478192717
// MI455X (gfx1250) — compile-verified
//
#include <hip/hip_runtime.h>

typedef unsigned int v4u __attribute__((ext_vector_type(4)));
typedef int          v4i __attribute__((ext_vector_type(4)));
typedef int          v8i __attribute__((ext_vector_type(8)));
typedef float        v2f __attribute__((ext_vector_type(2)));
typedef float        v4f __attribute__((ext_vector_type(4)));

#define T_STEPS 2048
#define B_DIM   512
#define N_DIM   128
#define BETA    0.9f
#define WAVES_PER_BLOCK 2     // 64 threads/block, 256 blocks -> 512 waves (one per b)

// Each wave owns one batch element b:
//   - TDM-gathers spike_seq[:, b, 0:2] (2048 rows x 8B, stride 4KB) into its LDS slice
//   - each of the 32 lanes owns 4 consecutive neurons n, runs the LIF scan,
//     and emits one nontemporal b128 store per timestep (coalesced 512B/wave).
__global__ __launch_bounds__(WAVES_PER_BLOCK * 32)
void FMFMNeuronBank_kernel(const float* __restrict__ spike,
                           const float* __restrict__ W,
                           float* __restrict__ out)
{
    __shared__ v2f sbuf[WAVES_PER_BLOCK][T_STEPS];   // 32 KB: [wave][t] = (s1, s2)

    const int tid  = threadIdx.x;
    const int lane = tid & 31;
    const int wv   = __builtin_amdgcn_readfirstlane(tid >> 5);
    const int b    = blockIdx.x * WAVES_PER_BLOCK + wv;

    // ---- Build Tensor DMA descriptor (D#) for this wave's gather ----
    // LDS byte offset = low 32 bits of the flat shared-aperture address.
    unsigned int lds_off =
        __builtin_amdgcn_readfirstlane((unsigned int)(unsigned long long)(&sbuf[wv][0]));
    unsigned long long gaddr = (unsigned long long)(spike + (size_t)b * 2);

    // Group 0: [1:0]=count=1, [63:32]=lds_addr, [120:64]=global_addr, [127:126]=type=2
    v4u g0;
    g0.x = 1u;
    g0.y = lds_off;
    g0.z = (unsigned int)(gaddr & 0xFFFFFFFFu);
    g0.w = ((unsigned int)(gaddr >> 32) & 0x01FFFFFFu) | (2u << 30);

    // Group 1: view memory as 2D tensor of 8-byte elements:
    //   dim0 (x) = B elements, dim0_stride = B elements (row = one timestep)
    //   dim1 (y) = T rows. Tile = 1 x T starting at column b (folded into global_addr).
    v8i g1;
    g1[0] = (int)(3u << 16);                 // workgroup_mask=0, data_size=3 (8B)
    g1[1] = (int)((unsigned)B_DIM << 16);    // tensor_dim0[15:0] @ bits 63:48
    g1[2] = (int)((unsigned)T_STEPS << 16);  // tensor_dim0 hi=0 | tensor_dim1[15:0]
    g1[3] = (int)(1u << 16);                 // tensor_dim1 hi=0 | tile_dim0 = 1
    g1[4] = T_STEPS;                         // tile_dim1 = 2048, tile_dim2 = 0
    g1[5] = B_DIM;                           // tensor_dim0_stride low 32 (elements)
    g1[6] = 0;                               // stride hi | tensor_dim1_stride low
    g1[7] = 0;

    // Groups 2/3: unused dims kept at extent 1 so OOB logic can't clip the tile.
    v4i g2 = {1, 1, 0, 0};                   // tensor_dim2=1, tensor_dim3=1
    v4i g3 = {0, 0x00010000, 0, 0};          // tensor_dim4=1, tile_dim4=0

#if __clang_major__ >= 23
    v8i gpad = {0, 0, 0, 0, 0, 0, 0, 0};
    __builtin_amdgcn_tensor_load_to_lds(g0, g1, g2, g3, gpad, 0);
#else
    __builtin_amdgcn_tensor_load_to_lds(g0, g1, g2, g3, 0);
#endif

    // ---- Per-lane neuron weights (4 neurons per lane) ----
    const int n0 = lane * 4;
    float w1[4], w2[4], mem[4], spk[4];
#pragma unroll
    for (int j = 0; j < 4; ++j) {
        w1[j]  = W[(n0 + j) * 2 + 0];
        w2[j]  = W[(n0 + j) * 2 + 1];
        mem[j] = 0.0f;
        spk[j] = 0.0f;
    }

    // Wait for the TDM gather to land in our private LDS slice.
    __builtin_amdgcn_s_wait_tensorcnt(0);

    float* outp = out + (size_t)b * N_DIM + n0;
    const v2f* sp = &sbuf[wv][0];

    // ---- Sequential LIF scan: mem = beta*mem + cur - spk_prev; spk = (mem > 1) ----
#pragma unroll 4
    for (int t = 0; t < T_STEPS; ++t) {
        v2f s = sp[t];                       // ds_load_b64, broadcast across wave
        v4f o;
#pragma unroll
        for (int j = 0; j < 4; ++j) {
            float cur = fmaf(s.x, w1[j], s.y * w2[j]);
            mem[j] = fmaf(BETA, mem[j], cur - spk[j]);
            spk[j] = (mem[j] > 1.0f) ? 1.0f : 0.0f;
            o[j] = spk[j];
        }
        // 512 MB write stream: nontemporal b128, don't pollute L2.
        __builtin_nontemporal_store(o, (v4f*)(outp + (size_t)t * (B_DIM * N_DIM)));
    }
}

extern "C" void kernel_launch(void* const* d_in, const int* in_sizes, int n_in,
                              void* d_out, int out_size, void* d_ws, size_t ws_size,
                              hipStream_t stream) {
    const float* spike = (const float*)d_in[0];   // [T, B, 2] fp32
    const float* W     = (const float*)d_in[1];   // [N, 2]    fp32
    float*       out   = (float*)d_out;           // [T, B, N] fp32

    dim3 grid(B_DIM / WAVES_PER_BLOCK);           // 256 blocks
    dim3 block(WAVES_PER_BLOCK * 32);             // 64 threads (2 waves)
    hipLaunchKernelGGL(FMFMNeuronBank_kernel, grid, block, 0, stream, spike, W, out);
}